// DisentangledStateEncoder_52673478918565
// MI455X (gfx1250) — compile-verified
//
#include <hip/hip_runtime.h>
#include <cstdint>
#include <cstddef>

#define B_ 256
#define S_ 1024
#define D_ 128
#define H_ 128
#define SD_ 5

typedef __attribute__((ext_vector_type(2))) float v2f;
typedef __attribute__((ext_vector_type(8))) float v8f;

__device__ __forceinline__ float gelu_exact(float v) {
  return 0.5f * v * (1.0f + erff(v * 0.70710678118654752440f));
}
__device__ __forceinline__ float sigm(float v) { return 1.0f / (1.0f + expf(-v)); }

// ---------------- Phase 1: h = gelu(LN(x@W1+b1)), bx = h@Winn+binn ----------------
// One block = 128 tokens, 8 waves; each wave owns a 16x128 strip via f32 WMMA.
__global__ __launch_bounds__(256) void dse_phase1(
    const float* __restrict__ x, const float* __restrict__ W1,
    const float* __restrict__ b1, const float* __restrict__ ln_g,
    const float* __restrict__ ln_b, const float* __restrict__ Winn,
    const float* __restrict__ binn,
    float* __restrict__ h_ws, float* __restrict__ bx_ws)
{
  __shared__ float smem[16384]; // 64KB: first W1 (packed), then reused as h tile
  const int tid  = threadIdx.x;
  const int lane = tid & 31, w = tid >> 5;
  const int hi = lane >> 4, l15 = lane & 15;
  const int tok0 = blockIdx.x * 128;
  const int b  = tok0 / S_;
  const int t0 = tok0 - b * S_;

  // Pack W1 so each B-fragment is one aligned 8B LDS load:
  // smem[(k>>1)*256 + n*2 + (k&1)] = W1[k][n]
  for (int idx = tid; idx < H_ * H_; idx += 256) {
    int k = idx >> 7, n = idx & 127;
    smem[(k >> 1) * 256 + n * 2 + (k & 1)] = W1[idx];
  }
  __syncthreads();

  const int mrow = tok0 + w * 16 + l15; // this lane's token row for A fragments
  v8f acc[8];
  #pragma unroll
  for (int j = 0; j < 8; ++j) {
    float bv = b1[j * 16 + l15];
    #pragma unroll
    for (int v = 0; v < 8; ++v) acc[j][v] = bv;
  }
  for (int k0 = 0; k0 < D_; k0 += 4) {
    const int kk = k0 + 2 * hi;
    v2f a = *(const v2f*)(x + (size_t)mrow * D_ + kk); // A 16x4 frag (row-major x)
    v2f bf[8];
    #pragma unroll
    for (int j = 0; j < 8; ++j)
      bf[j] = *(const v2f*)(smem + (kk >> 1) * 256 + (j * 16 + l15) * 2);
    #pragma unroll
    for (int j = 0; j < 8; ++j)
      acc[j] = __builtin_amdgcn_wmma_f32_16x16x4_f32(false, a, false, bf[j],
                                                     (short)0, acc[j], false, false);
  }

  // LayerNorm stats: lane holds rows (v + 8*hi); reduce over 8 j-tiles + 16 lanes.
  float mu[8], rs[8];
  #pragma unroll
  for (int v = 0; v < 8; ++v) {
    float s = 0.f, q = 0.f;
    #pragma unroll
    for (int j = 0; j < 8; ++j) { float t = acc[j][v]; s += t; q += t * t; }
    #pragma unroll
    for (int msk = 1; msk < 16; msk <<= 1) {
      s += __shfl_xor(s, msk, 32);
      q += __shfl_xor(q, msk, 32);
    }
    float m = s * (1.0f / 128.0f);
    mu[v] = m;
    rs[v] = rsqrtf(q * (1.0f / 128.0f) - m * m + 1e-5f);
  }

  __syncthreads(); // all waves done reading packed W1 -> reuse smem as h tile
  #pragma unroll
  for (int j = 0; j < 8; ++j) {
    int n = j * 16 + l15;
    float g = ln_g[n], bb = ln_b[n];
    #pragma unroll
    for (int v = 0; v < 8; ++v) {
      float val = (acc[j][v] - mu[v]) * rs[v] * g + bb;
      smem[(w * 16 + v + 8 * hi) * 128 + n] = gelu_exact(val);
    }
  }
  __syncthreads();

  // Store h in scan-major layout (S, B, H) for phase 2; compute bx.
  for (int idx = tid; idx < 128 * H_; idx += 256) {
    int r = idx >> 7, c = idx & 127;
    h_ws[((size_t)(t0 + r) * B_ + b) * H_ + c] = smem[idx];
  }
  for (int o = tid; o < 128 * SD_; o += 256) {
    int r = o / SD_, jj = o - r * SD_;
    float a2 = binn[jj];
    #pragma unroll 4
    for (int kk = 0; kk < H_; ++kk) {
      int k = (kk + r * 4) & 127; // skew to spread LDS banks
      a2 += smem[r * 128 + k] * Winn[k * SD_ + jj];
    }
    bx_ws[((size_t)(t0 + r) * B_ + b) * SD_ + jj] = a2;
  }
}

// ---------------- Phase 2: sequential scan, 16 blocks x 16 batch rows ----------------
// corr_in column map (K = 136): cols 0..4 = s_lin (Wc1 rows 0..4), cols 5..7 = zero,
// cols 8..135 = h (Wc1 rows 5..132). Col 8 start keeps async b128 LDS stores 16B-aligned.
__global__ __launch_bounds__(256) void dse_phase2(
    const float* __restrict__ h_ws, const float* __restrict__ bx_ws,
    const float* __restrict__ Wc1, const float* __restrict__ bc1,
    const float* __restrict__ Wc2, const float* __restrict__ bc2,
    const float* __restrict__ corr_scale,
    const float* __restrict__ raw_aL, const float* __restrict__ raw_aT,
    const float* __restrict__ raw_g, const float* __restrict__ raw_aR,
    const float* __restrict__ omega, float* __restrict__ out)
{
  __shared__ float cin[2][16 * 140];
  __shared__ float u_s[16 * 132];
  __shared__ float wc2_s[H_ * SD_];
  __shared__ float s_s[16 * 8];
  __shared__ float slin_s[16 * SD_];

  const int tid  = threadIdx.x;
  const int lane = tid & 31, w = tid >> 5;
  const int hi = lane >> 4, l15 = lane & 15;
  const int r0 = blockIdx.x * 16;

  const float a_l = sigm(raw_aL[0]) * 0.15f + 0.85f;
  const float a_t = sigm(raw_aT[0]) * 0.25f + 0.70f;
  const float gg  = sigm(raw_g[0])  * 0.20f + 0.80f;
  const float a_r = sigm(raw_aR[0]) * 0.40f;
  const float om  = omega[0];
  const float cw = cosf(om), sw = sinf(om);
  const float cs = corr_scale[0];

  for (int i = tid; i < 2 * 16 * 140; i += 256) (&cin[0][0])[i] = 0.f;
  for (int i = tid; i < 16 * 8; i += 256) s_s[i] = 0.f;
  for (int i = tid; i < H_ * SD_; i += 256) wc2_s[i] = Wc2[i];

  // Register-cache this wave's Wc1 B-fragments (N-tile w) with the column remap.
  const int n = w * 16 + l15;
  v2f bw[34];
  #pragma unroll
  for (int ki = 0; ki < 34; ++ki) {
    int kk = ki * 4 + 2 * hi;
    #pragma unroll
    for (int e = 0; e < 2; ++e) {
      int kc = kk + e;                        // corr_in column 0..135
      int wrow = (kc < 5) ? kc : (kc - 3);    // Wc1 row (valid when kc<5 or kc>=8)
      bool valid = (kc < 5) || (kc >= 8);
      float f = Wc1[(valid ? wrow : 0) * H_ + n];
      bw[ki][e] = valid ? f : 0.f;
    }
  }
  const float bc1v = bc1[n];
  const int m80 = tid / SD_, j80 = tid - m80 * SD_;
  const float bc2v = (tid < 80) ? bc2[j80] : 0.f;

  __syncthreads();

  // Async prefetch h tile for t=0 into buffer 0: 2 rows/wave, b128 per lane.
  #pragma unroll
  for (int q = 0; q < 2; ++q) {
    int m = w * 2 + q;
    unsigned loff = (unsigned)(size_t)(const void*)&cin[0][m * 140 + 8 + 4 * lane];
    const float* gp = h_ws + ((size_t)0 * B_ + (r0 + m)) * H_ + 4 * lane;
    asm volatile("global_load_async_to_lds_b128 %0, %1, off"
                 :: "v"(loff), "v"(gp) : "memory");
  }

  int cur = 0;
  for (int t = 0; t < S_; ++t) {
    asm volatile("s_wait_asynccnt 0x0" ::: "memory");
    __syncthreads(); // h[t] resident in cin[cur]; s_s from t-1 visible; cin[cur^1] free

    if (t + 1 < S_) { // prefetch next step's h tile into the other buffer
      #pragma unroll
      for (int q = 0; q < 2; ++q) {
        int m = w * 2 + q;
        unsigned loff = (unsigned)(size_t)(const void*)&cin[cur ^ 1][m * 140 + 8 + 4 * lane];
        const float* gp = h_ws + ((size_t)(t + 1) * B_ + (r0 + m)) * H_ + 4 * lane;
        asm volatile("global_load_async_to_lds_b128 %0, %1, off"
                     :: "v"(loff), "v"(gp) : "memory");
      }
    }

    if (tid < 80) { // s_lin = A(s) + bx_t, into corr_in cols 0..4
      float bx = bx_ws[((size_t)t * B_ + (r0 + m80)) * SD_ + j80];
      float s0 = s_s[m80 * 8 + 0], s1 = s_s[m80 * 8 + 1], s2 = s_s[m80 * 8 + 2];
      float s3 = s_s[m80 * 8 + 3], s4 = s_s[m80 * 8 + 4];
      float sv;
      if      (j80 == 0) sv = s0 * a_l;
      else if (j80 == 1) sv = s1 * a_t;
      else if (j80 == 2) sv = gg * ( cw * s2 + sw * s3);
      else if (j80 == 3) sv = gg * (-sw * s2 + cw * s3);
      else               sv = s4 * a_r;
      sv += bx;
      cin[cur][m80 * 140 + j80] = sv;
      slin_s[m80 * SD_ + j80] = sv;
    }
    __syncthreads();

    // GEMM1: (16x136) @ (136x16) per wave, B frags live in registers.
    v8f acc;
    #pragma unroll
    for (int v = 0; v < 8; ++v) acc[v] = bc1v;
    const float* cinp = &cin[cur][l15 * 140];
    #pragma unroll
    for (int ki = 0; ki < 34; ++ki) {
      v2f a = *(const v2f*)(cinp + ki * 4 + 2 * hi);
      acc = __builtin_amdgcn_wmma_f32_16x16x4_f32(false, a, false, bw[ki],
                                                  (short)0, acc, false, false);
    }
    #pragma unroll
    for (int v = 0; v < 8; ++v)
      u_s[(v + 8 * hi) * 132 + n] = gelu_exact(acc[v]);
    __syncthreads();

    if (tid < 80) { // GEMM2 (128 -> 5) + tanh correction + state update
      float mlp = bc2v;
      #pragma unroll 8
      for (int kk = 0; kk < H_; ++kk) {
        int k = (kk + m80 * 8) & 127; // bank skew
        mlp += u_s[m80 * 132 + k] * wc2_s[k * SD_ + j80];
      }
      s_s[m80 * 8 + j80] = slin_s[m80 * SD_ + j80] + cs * tanhf(mlp);
    }
    cur ^= 1;
  }
  __syncthreads();
  if (tid < 80) out[(r0 + m80) * SD_ + j80] = s_s[m80 * 8 + j80];
}

extern "C" void kernel_launch(void* const* d_in, const int* in_sizes, int n_in,
                              void* d_out, int out_size, void* d_ws, size_t ws_size,
                              hipStream_t stream) {
  (void)in_sizes; (void)n_in; (void)out_size; (void)ws_size;
  const float* x    = (const float*)d_in[0];
  const float* W1   = (const float*)d_in[1];
  const float* b1   = (const float*)d_in[2];
  const float* ln_g = (const float*)d_in[3];
  const float* ln_b = (const float*)d_in[4];
  const float* Winn = (const float*)d_in[5];
  const float* binn = (const float*)d_in[6];
  const float* Wc1  = (const float*)d_in[7];
  const float* bc1  = (const float*)d_in[8];
  const float* Wc2  = (const float*)d_in[9];
  const float* bc2  = (const float*)d_in[10];
  const float* csc  = (const float*)d_in[11];
  const float* raL  = (const float*)d_in[12];
  const float* raT  = (const float*)d_in[13];
  const float* rg   = (const float*)d_in[14];
  const float* raR  = (const float*)d_in[15];
  const float* omg  = (const float*)d_in[16];

  float* h_ws  = (float*)d_ws;                       // (S, B, H) f32 = 128 MB
  float* bx_ws = h_ws + (size_t)S_ * B_ * H_;        // (S, B, SD) f32 = 5 MB

  dse_phase1<<<(B_ * S_) / 128, 256, 0, stream>>>(x, W1, b1, ln_g, ln_b, Winn, binn,
                                                  h_ws, bx_ws);
  dse_phase2<<<B_ / 16, 256, 0, stream>>>(h_ws, bx_ws, Wc1, bc1, Wc2, bc2, csc,
                                          raL, raT, rg, raR, omg, (float*)d_out);
}